// ReprogrammingLayer_75471165325424
// MI455X (gfx1250) — compile-verified
//
#include <hip/hip_runtime.h>
#include <hip/hip_bf16.h>

// ---------------------------------------------------------------------------
// ReprogrammingLayer (Time-LLM style cross attention), fp32, CDNA5 / gfx1250.
// All GEMM work runs on V_WMMA_F32_16X16X4_F32 (full fp32 matrix pipe).
// B=8, L=1024, d_model=1024, S=1000 (padded to 1024), d_llm=4096, H=16, E=64.
// ---------------------------------------------------------------------------

typedef __attribute__((ext_vector_type(2))) float v2f;
typedef __attribute__((ext_vector_type(4))) float v4f;
typedef __attribute__((ext_vector_type(8))) float v8f;

__device__ __forceinline__ v8f wmma_f32_16x16x4(v8f acc, v2f a, v2f b) {
  // 8 args: (neg_a, A, neg_b, B, c_mod, C, reuse_a, reuse_b)
  return __builtin_amdgcn_wmma_f32_16x16x4_f32(
      /*neg_a=*/false, a, /*neg_b=*/false, b,
      /*c_mod=*/(short)0, acc, /*reuse_a=*/false, /*reuse_b=*/false);
}

// ---------------------------------------------------------------------------
// Generic C[M,N] = (COMBINE ? alpha*A1 + beta*A2 : A1)[M,K] @ W[K,N] + bias[N]
// Block: 256 threads = 8 waves. Block tile: 32 (M) x 256 (N). K-step: 16.
// Wave w owns rows 16*(w>>2)..+16 and four 16-col N-subtiles at 64*(w&3):
// 4 accumulators per wave, A fragment reused 4x per K-substep.
// Rows in [M, Mstore) are written as zero (used to zero-pad K/V to S=1024).
// Requires N % 256 == 0, K % 16 == 0.
// ---------------------------------------------------------------------------
template <bool COMBINE>
__global__ __launch_bounds__(256) void gemm_bias_f32(
    const float* __restrict__ A1, const float* __restrict__ A2,
    const float* __restrict__ alphaP, const float* __restrict__ betaP,
    const float* __restrict__ W, const float* __restrict__ bias,
    float* __restrict__ C, int M, int K, int N, int Mstore) {
  __shared__ float sA[32][16];    //  2 KB
  __shared__ float sW[16][256];   // 16 KB

  const int tid  = threadIdx.x;
  const int wave = tid >> 5;
  const int lane = tid & 31;
  const int half = lane >> 4;     // K-half selector for A/B fragments
  const int idx  = lane & 15;     // A: row, B: col

  const int mh = wave >> 2;       // 16-row half of the 32-row tile
  const int nq = wave & 3;        // 64-col quarter of the 256-col tile

  const int m0 = blockIdx.y * 32;
  const int n0 = blockIdx.x * 256;

  float alpha = 1.0f, beta = 0.0f;
  if (COMBINE) { alpha = *alphaP; beta = *betaP; }

  v8f acc0 = {0.f,0.f,0.f,0.f,0.f,0.f,0.f,0.f};
  v8f acc1 = acc0, acc2 = acc0, acc3 = acc0;

  for (int k0 = 0; k0 < K; k0 += 16) {
    // Stage A tile: 32x16 (2 elements / thread, coalesced within rows)
    for (int i = tid; i < 32 * 16; i += 256) {
      int r = i >> 4, c = i & 15;
      int gr = m0 + r;
      float a = 0.0f;
      if (gr < M) {
        size_t off = (size_t)gr * K + k0 + c;
        a = A1[off];
        if (COMBINE) a = alpha * a + beta * A2[off];
      }
      sA[r][c] = a;
    }
    // Stage W tile: 16x256 as float4 (4 vec loads / thread, coalesced)
    {
      const float* wrow = W + (size_t)k0 * N + n0;
      for (int i = tid * 4; i < 16 * 256; i += 256 * 4) {
        int r = i >> 8, c = i & 255;
        v4f w = *(const v4f*)(wrow + (size_t)r * N + c);
        *(v4f*)&sW[r][c] = w;
      }
    }
    __syncthreads();

#pragma unroll
    for (int ks = 0; ks < 16; ks += 4) {
      // A 16x4 fragment: lane holds A[idx][ks + 2*half + j]; reused 4x
      v2f a;
      a[0] = sA[mh * 16 + idx][ks + 2 * half + 0];
      a[1] = sA[mh * 16 + idx][ks + 2 * half + 1];
      // B 4x16 fragments: lane holds W[ks + 2*half + j][col]
      const int cb = nq * 64 + idx;
      v2f b0, b1, b2, b3;
      b0[0] = sW[ks + 2*half + 0][cb +  0]; b0[1] = sW[ks + 2*half + 1][cb +  0];
      b1[0] = sW[ks + 2*half + 0][cb + 16]; b1[1] = sW[ks + 2*half + 1][cb + 16];
      b2[0] = sW[ks + 2*half + 0][cb + 32]; b2[1] = sW[ks + 2*half + 1][cb + 32];
      b3[0] = sW[ks + 2*half + 0][cb + 48]; b3[1] = sW[ks + 2*half + 1][cb + 48];
      acc0 = wmma_f32_16x16x4(acc0, a, b0);
      acc1 = wmma_f32_16x16x4(acc1, a, b1);
      acc2 = wmma_f32_16x16x4(acc2, a, b2);
      acc3 = wmma_f32_16x16x4(acc3, a, b3);
    }
    __syncthreads();
  }

  // C/D layout: VGPR v, lanes 0-15 -> M=v, lanes 16-31 -> M=v+8
  v8f accs[4] = {acc0, acc1, acc2, acc3};
#pragma unroll
  for (int sub = 0; sub < 4; sub++) {
    const int col = n0 + nq * 64 + sub * 16 + idx;
    const float bv = bias[col];
#pragma unroll
    for (int v = 0; v < 8; v++) {
      int row = m0 + mh * 16 + v + 8 * half;
      if (row < M)
        C[(size_t)row * N + col] = accs[sub][v] + bv;
      else if (row < Mstore)
        C[(size_t)row * N + col] = 0.0f;   // zero the S-padding rows
    }
  }
}

// ---------------------------------------------------------------------------
// Attention for one (b, h, 32-row L tile). 256 threads = 8 waves.
// Dynamic LDS: scores[32][Sp] (128KB) + Qtile[32][64] + red[32][8].
//   phase 1: scores = (Q Kt) * 1/sqrt(E) via WMMA, mask cols >= S to -inf
//   phase 2: row softmax in LDS (8 threads per row)
//   phase 3: O = P V via WMMA (8 waves = 8 disjoint 16x16 output tiles)
// ---------------------------------------------------------------------------
#define RT 32              // L rows per block
#define DIN 1024           // H*E

__global__ __launch_bounds__(256) void attn_kernel(
    const float* __restrict__ Q,   // [B*L, DIN]
    const float* __restrict__ Km,  // [Sp, DIN]  (pad rows zeroed)
    const float* __restrict__ Vm,  // [Sp, DIN]  (pad rows zeroed)
    float* __restrict__ O,         // [B*L, DIN]
    int Bb, int L, int S, int Sp) {
  extern __shared__ float smem[];
  float* sc  = smem;                 // [RT][Sp]
  float* qld = sc + RT * Sp;         // [RT][64]
  float* red = qld + RT * 64;        // [RT][8]

  const int ltiles = L / RT;
  const int bid = blockIdx.x;
  const int l0 = (bid % ltiles) * RT;
  const int h  = (bid / ltiles) % 16;
  const int b  = bid / (ltiles * 16);

  const int tid  = threadIdx.x;
  const int wave = tid >> 5;
  const int lane = tid & 31;
  const int half = lane >> 4;
  const int idx  = lane & 15;

  const size_t qrow0 = (size_t)b * L + l0;

  // Load Q tile (32x64), coalesced
  for (int i = tid; i < RT * 64; i += 256) {
    int r = i >> 6, e = i & 63;
    qld[r * 64 + e] = Q[(qrow0 + r) * DIN + h * 64 + e];
  }
  __syncthreads();

  const float scale = 0.125f;  // 1/sqrt(64)
  const float NEGINF = -__builtin_inff();

  // -------- phase 1: scores --------
  // wave -> row half (wave&1), strided S-tiles (wave>>1)
  {
    const int rh = wave & 1;
    for (int st = (wave >> 1); st < Sp / 16; st += 4) {
      v8f acc = {0.f,0.f,0.f,0.f,0.f,0.f,0.f,0.f};
      const float* kp = Km + (size_t)(st * 16 + idx) * DIN + h * 64;
#pragma unroll
      for (int ks = 0; ks < 64; ks += 4) {
        v2f a, bf;
        a[0]  = qld[(rh * 16 + idx) * 64 + ks + 2 * half + 0];
        a[1]  = qld[(rh * 16 + idx) * 64 + ks + 2 * half + 1];
        bf[0] = kp[ks + 2 * half + 0];   // Kt fragment: B[kk][n] = K[s][kk]
        bf[1] = kp[ks + 2 * half + 1];
        acc = wmma_f32_16x16x4(acc, a, bf);
      }
#pragma unroll
      for (int v = 0; v < 8; v++) {
        int r = rh * 16 + v + 8 * half;
        int s = st * 16 + idx;
        sc[r * Sp + s] = (s < S) ? acc[v] * scale : NEGINF;
      }
    }
  }
  __syncthreads();

  // -------- phase 2: softmax (8 threads per row, 128 cols each) --------
  {
    const int r = tid >> 3;
    const int seg = tid & 7;
    float* rowp = sc + r * Sp;
    const int c0 = seg * (Sp / 8);
    const int c1 = c0 + (Sp / 8);

    float mx = NEGINF;
    for (int s = c0; s < c1; s++) mx = fmaxf(mx, rowp[s]);
    red[r * 8 + seg] = mx;
    __syncthreads();
    mx = red[r * 8 + 0];
#pragma unroll
    for (int j = 1; j < 8; j++) mx = fmaxf(mx, red[r * 8 + j]);
    __syncthreads();

    float sum = 0.0f;
    for (int s = c0; s < c1; s++) {
      float p = __expf(rowp[s] - mx);   // exp(-inf - mx) == 0 for padding
      rowp[s] = p;
      sum += p;
    }
    red[r * 8 + seg] = sum;
    __syncthreads();
    float tot = 0.0f;
#pragma unroll
    for (int j = 0; j < 8; j++) tot += red[r * 8 + j];
    const float inv = 1.0f / tot;
    for (int s = c0; s < c1; s++) rowp[s] *= inv;
  }
  __syncthreads();

  // -------- phase 3: O = P @ V --------
  {
    const int rh = wave >> 2;      // row half
    const int nt = wave & 3;       // 16-col E subtile
    v8f acc = {0.f,0.f,0.f,0.f,0.f,0.f,0.f,0.f};
#pragma unroll 4
    for (int s0 = 0; s0 < Sp; s0 += 4) {
      v2f a, bf;
      a[0]  = sc[(rh * 16 + idx) * Sp + s0 + 2 * half + 0];
      a[1]  = sc[(rh * 16 + idx) * Sp + s0 + 2 * half + 1];
      bf[0] = Vm[(size_t)(s0 + 2 * half + 0) * DIN + h * 64 + nt * 16 + idx];
      bf[1] = Vm[(size_t)(s0 + 2 * half + 1) * DIN + h * 64 + nt * 16 + idx];
      acc = wmma_f32_16x16x4(acc, a, bf);
    }
#pragma unroll
    for (int v = 0; v < 8; v++) {
      int r = rh * 16 + v + 8 * half;
      int e = nt * 16 + idx;
      O[(qrow0 + r) * DIN + h * 64 + e] = acc[v];
    }
  }
}

// ---------------------------------------------------------------------------
// Host-side orchestration (graph-capture safe; all launches on `stream`).
// ---------------------------------------------------------------------------
extern "C" void kernel_launch(void* const* d_in, const int* in_sizes, int n_in,
                              void* d_out, int out_size, void* d_ws, size_t ws_size,
                              hipStream_t stream) {
  (void)in_sizes; (void)n_in; (void)out_size; (void)ws_size;

  const float* tgt   = (const float*)d_in[0];   // [8,1024,1024]
  const float* src   = (const float*)d_in[1];   // [1000,4096]
  const float* val   = (const float*)d_in[2];   // [1000,4096]
  const float* Wq_s  = (const float*)d_in[3];
  const float* bq_s  = (const float*)d_in[4];
  const float* Wk_s  = (const float*)d_in[5];
  const float* bk_s  = (const float*)d_in[6];
  const float* Wv_s  = (const float*)d_in[7];
  const float* bv_s  = (const float*)d_in[8];
  const float* Wq_l  = (const float*)d_in[9];
  const float* bq_l  = (const float*)d_in[10];
  const float* Wk_l  = (const float*)d_in[11];
  const float* bk_l  = (const float*)d_in[12];
  const float* Wv_l  = (const float*)d_in[13];
  const float* bv_l  = (const float*)d_in[14];
  const float* Wo    = (const float*)d_in[15];  // [1024,4096]
  const float* bo    = (const float*)d_in[16];  // [4096]
  const float* alpha = (const float*)d_in[17];
  const float* beta  = (const float*)d_in[18];

  const int B = 8, L = 1024, Dm = 1024, Dllm = 4096, S = 1000, Sp = 1024;
  const int BL = B * L;            // 8192
  const int Din = 1024;            // H*E

  // Workspace layout (fp32): ~151 MB
  float* ws  = (float*)d_ws;
  const size_t QN = (size_t)BL * Din;   // 8M floats
  const size_t KN = (size_t)Sp * Din;   // 1M floats
  float* q_s = ws;
  float* q_l = q_s + QN;
  float* k_s = q_l + QN;
  float* v_s = k_s + KN;
  float* k_l = v_s + KN;
  float* v_l = k_l + KN;
  float* a_s = v_l + KN;
  float* a_l = a_s + QN;

  dim3 blk(256);

  // Q projections: [8192,1024] @ [1024,1024]
  {
    dim3 grd(Din / 256, BL / 32);
    gemm_bias_f32<false><<<grd, blk, 0, stream>>>(tgt, nullptr, nullptr, nullptr,
                                                  Wq_s, bq_s, q_s, BL, Dm, Din, BL);
    gemm_bias_f32<false><<<grd, blk, 0, stream>>>(tgt, nullptr, nullptr, nullptr,
                                                  Wq_l, bq_l, q_l, BL, Dm, Din, BL);
  }
  // K/V projections: [1000,4096] @ [4096,1024] -> padded to 1024 rows (zeros)
  {
    dim3 grd(Din / 256, Sp / 32);
    gemm_bias_f32<false><<<grd, blk, 0, stream>>>(src, nullptr, nullptr, nullptr,
                                                  Wk_s, bk_s, k_s, S, Dllm, Din, Sp);
    gemm_bias_f32<false><<<grd, blk, 0, stream>>>(val, nullptr, nullptr, nullptr,
                                                  Wv_s, bv_s, v_s, S, Dllm, Din, Sp);
    gemm_bias_f32<false><<<grd, blk, 0, stream>>>(src, nullptr, nullptr, nullptr,
                                                  Wk_l, bk_l, k_l, S, Dllm, Din, Sp);
    gemm_bias_f32<false><<<grd, blk, 0, stream>>>(val, nullptr, nullptr, nullptr,
                                                  Wv_l, bv_l, v_l, S, Dllm, Din, Sp);
  }
  // Attention (both branches)
  {
    const int nblk = B * 16 * (L / RT);                       // 4096
    const size_t shmem = (size_t)(RT * Sp + RT * 64 + RT * 8) * sizeof(float); // ~137KB
    attn_kernel<<<nblk, 256, shmem, stream>>>(q_s, k_s, v_s, a_s, B, L, S, Sp);
    attn_kernel<<<nblk, 256, shmem, stream>>>(q_l, k_l, v_l, a_l, B, L, S, Sp);
  }
  // Output projection: (alpha*a_s + beta*a_l) @ Wo + bo -> [8192,4096]
  {
    dim3 grd(Dllm / 256, BL / 32);
    gemm_bias_f32<true><<<grd, blk, 0, stream>>>(a_s, a_l, alpha, beta,
                                                 Wo, bo, (float*)d_out,
                                                 BL, Din, Dllm, BL);
  }
}